// SS3T_9972914061498
// MI455X (gfx1250) — compile-verified
//
#include <hip/hip_runtime.h>
#include <math.h>

// ---------------- problem constants ----------------
#define N_SH    45
#define NP      48        // padded param count (47 real + 1 pad, pad stays 0)
#define NPAD    52        // LDS row width for param arrays (max read idx = 49)
#define NMEAS   64
#define VPB     8         // voxels per block
#define VPBP    16        // padded voxel rows (lanes 0..15 load unguarded)
#define NCAND   64        // VPB * 8 alphas
#define NITERS  10
#define N_VOX   16384

#define RGM     0.3f
#define RCSF    0.1f
#define L1W     0.01f
#define LPRI    1.0f
#define INV_S2  400.0f    // 1 / (0.05^2)
#define SQRT4PI 3.5449077018110318f

typedef __attribute__((ext_vector_type(2))) float vf2_t;
typedef __attribute__((ext_vector_type(8))) float vf8_t;
typedef __attribute__((ext_vector_type(4))) unsigned int u32x4_t;
typedef __attribute__((ext_vector_type(4))) int i32x4_t;
typedef __attribute__((ext_vector_type(8))) int i32x8_t;

#if __has_builtin(__builtin_amdgcn_tensor_load_to_lds)
#define USE_TDM 1
#else
#define USE_TDM 0
#endif

// V_WMMA_F32_16X16X4_F32: D(16x16 f32) = A(16x4) * B(4x16) + C
static __device__ __forceinline__ vf8_t wmma4(vf2_t a, vf2_t b, vf8_t c) {
  return __builtin_amdgcn_wmma_f32_16x16x4_f32(false, a, false, b, (short)0, c,
                                               false, false);
}

// ---------------- shared state per block (8 voxels) ----------------
struct SM {
  float Yr[NMEAS][NP];        // SH basis * response, padded cols 45..47 = 0
  float dStage[VPB][NMEAS];   // TDM landing zone (row-major, as in HBM)
  float dT[NMEAS][VPBP];      // data transposed [meas][voxel], cols 8..15 pad
  float H[VPB][NP][NP];       // BFGS inverse-Hessian
  float P[VPBP][NPAD];        // current params   (pad rows/cols stay 0)
  float G[VPBP][NPAD];        // current grad
  float Gn[VPBP][NPAD];       // new grad scratch
  float Dv[VPBP][NPAD];       // search direction (pad rows/cols stay 0)
  float Sv[VPB][NP];          // s = p1 - p
  float Yv[VPB][NP];          // y = g1 - g
  float Hy[VPB][NP];          // H @ y
  float dLdI[NMEAS][VPBP];    // dLoss/d(inner signal), cols 8..15 junk
  float candLoss[NCAND];
  float fpart[VPBP], dpart[VPBP];
  float fcur[VPB], astep[VPB], rhov[VPB], yhyv[VPB];
  float pr[VPBP][3];
};

// ---------------- math helpers ----------------
static __device__ __forceinline__ float softplusf(float x) {
  return log1pf(expf(-fabsf(x))) + fmaxf(x, 0.0f);
}
static __device__ __forceinline__ float sigmoidf(float x) {
  return 1.0f / (1.0f + expf(-x));
}
// log(i0e(x)) for x >= 0  (Abramowitz & Stegun 9.8.1 / 9.8.2)
static __device__ __forceinline__ float log_i0e(float x) {
  if (x < 3.75f) {
    float t = x * (1.0f / 3.75f); t *= t;
    float p = 1.0f + t*(3.5156229f + t*(3.0899424f + t*(1.2067492f +
              t*(0.2659732f + t*(0.0360768f + t*0.0045813f)))));
    return __logf(p) - x;
  } else {
    float u = 3.75f / x;
    float p = 0.39894228f + u*(0.01328592f + u*(0.00225319f + u*(-0.00157565f +
              u*(0.00916281f + u*(-0.02057706f + u*(0.02635537f +
              u*(-0.01647633f + u*0.00392377f)))))));
    return __logf(p) - 0.5f * __logf(x);
  }
}
// I1(x)/I0(x) for x >= 0  (A&S 9.8.1-9.8.4)
static __device__ __forceinline__ float bessel_ratio(float x) {
  if (x < 3.75f) {
    float t = x * (1.0f / 3.75f); t *= t;
    float p0 = 1.0f + t*(3.5156229f + t*(3.0899424f + t*(1.2067492f +
               t*(0.2659732f + t*(0.0360768f + t*0.0045813f)))));
    float p1 = x * (0.5f + t*(0.87890594f + t*(0.51498869f + t*(0.15084934f +
               t*(0.02658733f + t*(0.00301532f + t*0.00032411f))))));
    return p1 / p0;
  } else {
    float u = 3.75f / x;
    float p0 = 0.39894228f + u*(0.01328592f + u*(0.00225319f + u*(-0.00157565f +
               u*(0.00916281f + u*(-0.02057706f + u*(0.02635537f +
               u*(-0.01647633f + u*0.00392377f)))))));
    float p1 = 0.39894228f + u*(-0.03988024f + u*(-0.00362018f + u*(0.00163801f +
               u*(-0.01031555f + u*(0.02282967f + u*(-0.02895312f +
               u*(0.01787654f - u*0.00420059f)))))));
    return p1 / p0;
  }
}
static __device__ __forceinline__ float rician_nll_elem(float d, float S) {
  float xr = d * S * INV_S2;
  return -(__logf(d * INV_S2) - (d * d + S * S) * (0.5f * INV_S2) +
           log_i0e(xr) + xr);
}
static __device__ __forceinline__ float rician_dnll_dS(float d, float S) {
  float xr = d * S * INV_S2;
  return S * INV_S2 - d * INV_S2 * bessel_ratio(xr);
}
static __device__ __forceinline__ float sq(float x) { return x * x; }

// ---------------- value + grad at Pm, batched over 8 voxels ----------------
// All WMMA operand loads are unguarded; padded rows/cols make them safe.
static __device__ void vg_eval(SM& sm, float (*Pm)[NPAD], float (*Gout)[NPAD],
                               float* fout) {
  const int tid  = threadIdx.x;
  const int wv   = tid >> 5;
  const int lane = tid & 31;
  const int lm   = lane & 15;
  const int hf   = lane >> 4;

  if (tid < VPBP) { sm.fpart[tid] = 0.0f; sm.dpart[tid] = 0.0f; }
  __syncthreads();

  // ---- forward GEMM: S_inner(64 x 16pad) = Yr(64x48) @ C(48x16pad) ----
  if (wv < 4) {
    const int mt = wv;
    const int m  = 16 * mt + lm;
    const int v  = lm;                       // columns 8..15 = pad voxels
    vf8_t acc = {};
    for (int ks = 0; ks < 12; ++ks) {
      int k0 = 4 * ks + 2 * hf;
      vf2_t a; a.x = sm.Yr[m][k0]; a.y = sm.Yr[m][k0 + 1];
      int kb0 = 4 * ks + hf, kb1 = kb0 + 2;
      vf2_t b; b.x = Pm[v][2 + kb0]; b.y = Pm[v][2 + kb1];
      acc = wmma4(a, b, acc);
    }
    float wg = softplusf(Pm[v][0]);
    float wc = softplusf(Pm[v][1]);
    float base = wg * RGM + wc * RCSF;
    float lsum = 0.0f, dsum = 0.0f;
    for (int i = 0; i < 8; ++i) {
      int mm = 16 * mt + i + 8 * hf;
      float Sin = acc[i] + base;
      float S   = fabsf(Sin);
      float sgn = (Sin >= 0.0f) ? 1.0f : -1.0f;
      float d   = sm.dT[mm][v];
      lsum += rician_nll_elem(d, S);
      float g = rician_dnll_dS(d, S) * sgn;
      dsum += g;
      sm.dLdI[mm][v] = g;
    }
    atomicAdd(&sm.fpart[v], lsum);
    atomicAdd(&sm.dpart[v], dsum);
  }
  __syncthreads();

  // ---- per-voxel scalar terms: f, and grad wrt w_gm/w_csf (pre-softplus) ----
  if (tid < VPB) {
    int v = tid;
    float wg = softplusf(Pm[v][0]), wc = softplusf(Pm[v][1]);
    float sabs = 0.0f;
    for (int k = 0; k < N_SH; ++k) sabs += fabsf(Pm[v][2 + k]);
    float c0 = Pm[v][2];
    fout[v] = sm.fpart[v] + L1W * sabs +
              LPRI * (sq(wg - sm.pr[v][0]) + sq(wc - sm.pr[v][1]) +
                      sq(c0 * SQRT4PI - sm.pr[v][2]));
  }
  if (tid < 2 * VPB) {
    int v = tid >> 1, i = tid & 1;
    float pv  = Pm[v][i];
    float w   = softplusf(pv);
    float rr  = i ? RCSF : RGM;
    float raw = rr * sm.dpart[v] + 2.0f * LPRI * (w - sm.pr[v][i]);
    Gout[v][i] = raw * sigmoidf(pv);
  }
  __syncthreads();

  // ---- backward GEMM: grad_c(48 x 16pad) = Yr^T(48x64) @ dLdI(64x16pad) ----
  if (wv < 3) {
    const int mt = wv;
    const int r  = 16 * mt + lm;             // param-row (c index)
    vf8_t acc = {};
    for (int ks = 0; ks < 16; ++ks) {
      int m0 = 4 * ks + 2 * hf;              // K = measurement
      vf2_t a; a.x = sm.Yr[m0][r]; a.y = sm.Yr[m0 + 1][r];
      int mb0 = 4 * ks + hf, mb1 = mb0 + 2;
      vf2_t b; b.x = sm.dLdI[mb0][lm]; b.y = sm.dLdI[mb1][lm];
      acc = wmma4(a, b, acc);
    }
    int v = lm;                              // rows 8..15 = pad voxels (junk ok)
    for (int i = 0; i < 8; ++i) {
      int kk = 16 * mt + i + 8 * hf;         // c index (up to 47 -> col 49 pad)
      float val = acc[i];
      if (kk < N_SH) {
        float c = Pm[v][2 + kk];
        val += L1W * ((c > 0.0f) ? 1.0f : ((c < 0.0f) ? -1.0f : 0.0f));
        if (kk == 0)
          val += 2.0f * LPRI * (c * SQRT4PI - sm.pr[v][2]) * SQRT4PI;
      }
      Gout[v][2 + kk] = val;
    }
  }
  __syncthreads();
}

// ---------------- main kernel: one block fits 8 voxels ----------------
__global__ __launch_bounds__(256) void ss3t_fit_kernel(
    const float* __restrict__ data, const float* __restrict__ bvecs,
    const float* __restrict__ priors, const float* __restrict__ resp,
    float* __restrict__ out) {
  __shared__ SM sm;
  const int tid  = threadIdx.x;
  const int wv   = tid >> 5;
  const int lane = tid & 31;
  const int lm   = lane & 15;
  const int hf   = lane >> 4;
  const int vb   = blockIdx.x * VPB;

#if USE_TDM
  // ---- Tensor Data Mover: stage 8x64 f32 data block (2KB) global -> LDS ----
  if (wv == 0) {
    unsigned long long ga =
        (unsigned long long)(const void*)(data + (size_t)vb * NMEAS);
    unsigned lds_addr = (unsigned)__builtin_offsetof(SM, dStage);
    // D# group0: count=1 | lds_addr | global_addr[56:0] | type=2
    u32x4_t g0 = {1u, lds_addr, (unsigned)(ga & 0xffffffffu),
                  (unsigned)((ga >> 32) & 0x01ffffffu) | (2u << 30)};
    // D# group1: data_size=4B; 1-D tensor/tile of 512 elements
    i32x8_t g1 = {(int)(2u << 16),      // wg_mask=0, data_size=2 (4 bytes)
                  (int)(512u << 16),    // tensor_dim0[15:0] = 512
                  (int)(1u << 16),      // tensor_dim0[31:16]=0, tensor_dim1=1
                  (int)(512u << 16),    // tensor_dim1 hi=0, tile_dim0 = 512
                  0,                    // tile_dim1=0 (unused), tile_dim2=0
                  512, 0, 0};           // tensor_dim0_stride = 512
    i32x4_t gz = {0, 0, 0, 0};
#if defined(__clang_major__) && (__clang_major__ >= 23)
    i32x8_t gz8 = {0, 0, 0, 0, 0, 0, 0, 0};
    __builtin_amdgcn_tensor_load_to_lds(g0, g1, gz, gz, gz8, 0);
#else
    __builtin_amdgcn_tensor_load_to_lds(g0, g1, gz, gz, 0);
#endif
    __builtin_amdgcn_s_wait_tensorcnt(0);
  }
#endif

  // ---- concurrent init: priors, H = I, P = p0, zero padded arrays ----
  if (tid < VPB * 3) {
    int v = tid / 3, j = tid % 3;
    sm.pr[v][j] = priors[(size_t)(vb + v) * 3 + j];
  }
  if (tid >= VPB * 3 && tid < VPBP * 3) {   // pad priors
    sm.pr[tid / 3][tid % 3] = 0.0f;
  }
  for (int idx = tid; idx < VPB * NP * NP; idx += 256) {
    int v = idx / (NP * NP), rc = idx % (NP * NP);
    sm.H[v][rc / NP][rc % NP] = (rc / NP == rc % NP) ? 1.0f : 0.0f;
  }
  for (int idx = tid; idx < VPBP * NPAD; idx += 256) {
    int v = idx / NPAD, r = idx % NPAD;
    float p0 = 0.0f;
    if (v < VPB && r < NP) p0 = (r < 2) ? -2.0f : ((r == 2) ? 0.1f : 0.0f);
    sm.P[v][r]  = p0;
    sm.G[v][r]  = 0.0f;
    sm.Gn[v][r] = 0.0f;
    sm.Dv[v][r] = 0.0f;
  }
#if !USE_TDM
  for (int idx = tid; idx < VPB * NMEAS; idx += 256) {
    int v = idx >> 6, m = idx & 63;
    sm.dStage[v][m] = data[(size_t)(vb + v) * NMEAS + m];
  }
#endif

  // ---- build Yr = SH basis * response (64 directions, order 8) ----
  if (tid < NMEAS) {
    float bx = bvecs[tid * 3 + 0], by = bvecs[tid * 3 + 1], bz = bvecs[tid * 3 + 2];
    float rn = sqrtf(bx * bx + by * by + bz * bz);
    double x = (double)(bz / rn);
    x = fmin(1.0, fmax(-1.0, x));
    double sx = sqrt(fmax(0.0, 1.0 - x * x));
    double ph = atan2((double)by, (double)bx);
    for (int k = 0; k < NP; ++k) sm.Yr[tid][k] = 0.0f;
    const int off2[5] = {0, 1, 6, 15, 28};   // column offsets for l=0,2,4,6,8
    double dfact = 1.0;   // (2m-1)!!
    double sxm   = 1.0;   // sx^m
    for (int m = 0; m <= 8; ++m) {
      double pmm  = ((m & 1) ? -1.0 : 1.0) * dfact * sxm;
      double Pl_2 = pmm;                      // P_{m,m}
      double Pl_1 = x * (2 * m + 1) * pmm;    // P_{m+1,m}
      for (int l = m; l <= 8; ++l) {
        double Plm;
        if (l == m)          Plm = Pl_2;
        else if (l == m + 1) Plm = Pl_1;
        else {
          Plm = ((2 * l - 1) * x * Pl_1 - (l + m - 1) * Pl_2) / (double)(l - m);
          Pl_2 = Pl_1; Pl_1 = Plm;
        }
        if ((l & 1) == 0) {
          double fr = 1.0;   // (l-m)!/(l+m)!
          for (int i2 = l - m + 1; i2 <= l + m; ++i2) fr /= (double)i2;
          double nrm = sqrt((2 * l + 1) / (4.0 * 3.14159265358979323846) * fr);
          int base = off2[l >> 1] + l;
          if (m == 0) {
            sm.Yr[tid][base] = (float)(nrm * Plm * (double)resp[base]);
          } else {
            double s2 = 1.4142135623730951;
            int cc = base + m, cs = base - m;
            sm.Yr[tid][cc] = (float)(s2 * nrm * Plm * cos(m * ph) * (double)resp[cc]);
            sm.Yr[tid][cs] = (float)(s2 * nrm * Plm * sin(m * ph) * (double)resp[cs]);
          }
        }
      }
      dfact *= (double)(2 * m + 1);
      sxm   *= sx;
    }
  }
  __syncthreads();

  // ---- transpose staged data into [meas][voxel]; pad voxels -> 1.0 ----
  for (int idx = tid; idx < VPB * NMEAS; idx += 256) {
    int v = idx >> 6, m = idx & 63;
    sm.dT[m][v] = sm.dStage[v][m];
  }
  for (int idx = tid; idx < VPB * NMEAS; idx += 256) {
    int v = VPB + (idx >> 6), m = idx & 63;
    sm.dT[m][v] = 1.0f;
  }

  // ---- initial f0, g0 (vg_eval starts with its own barrier) ----
  vg_eval(sm, sm.P, sm.G, sm.fcur);

  // ---- BFGS iterations ----
  for (int it = 0; it < NITERS; ++it) {
    // dvec = -(H @ g)
    for (int idx = tid; idx < VPB * NP; idx += 256) {
      int v = idx / NP, r = idx % NP;
      float acc = 0.0f;
      for (int j = 0; j < NP; ++j) acc += sm.H[v][r][j] * sm.G[v][j];
      sm.Dv[v][r] = -acc;
    }
    for (int idx = tid; idx < NCAND; idx += 256) sm.candLoss[idx] = 0.0f;
    __syncthreads();

    // ---- line-search GEMM: S(64 x 64cand) = Yr @ (p + alpha*d) ----
    for (int job = wv; job < 16; job += 8) {
      int mt = job >> 2, nt = job & 3;
      int n  = 16 * nt + lm, v = n >> 3, ai = n & 7;
      float alpha = exp2f(-(float)ai);
      vf8_t acc = {};
      for (int ks = 0; ks < 12; ++ks) {
        int m  = 16 * mt + lm;
        int k0 = 4 * ks + 2 * hf;
        vf2_t a; a.x = sm.Yr[m][k0]; a.y = sm.Yr[m][k0 + 1];
        int kb0 = 4 * ks + hf, kb1 = kb0 + 2;
        vf2_t b;
        b.x = sm.P[v][2 + kb0] + alpha * sm.Dv[v][2 + kb0];
        b.y = sm.P[v][2 + kb1] + alpha * sm.Dv[v][2 + kb1];
        acc = wmma4(a, b, acc);
      }
      float wg = softplusf(sm.P[v][0] + alpha * sm.Dv[v][0]);
      float wc = softplusf(sm.P[v][1] + alpha * sm.Dv[v][1]);
      float base = wg * RGM + wc * RCSF;
      float lsum = 0.0f;
      for (int i = 0; i < 8; ++i) {
        int m = 16 * mt + i + 8 * hf;
        float S = fabsf(acc[i] + base);
        lsum += rician_nll_elem(sm.dT[m][v], S);
      }
      atomicAdd(&sm.candLoss[n], lsum);
    }
    __syncthreads();

    // add regularizer/prior terms per candidate; non-finite -> +inf
    if (tid < NCAND) {
      int n = tid, v = n >> 3, ai = n & 7;
      float alpha = exp2f(-(float)ai);
      float sabs = 0.0f;
      for (int k = 0; k < N_SH; ++k)
        sabs += fabsf(sm.P[v][2 + k] + alpha * sm.Dv[v][2 + k]);
      float wg = softplusf(sm.P[v][0] + alpha * sm.Dv[v][0]);
      float wc = softplusf(sm.P[v][1] + alpha * sm.Dv[v][1]);
      float c0 = sm.P[v][2] + alpha * sm.Dv[v][2];
      float f  = sm.candLoss[n] + L1W * sabs +
                 LPRI * (sq(wg - sm.pr[v][0]) + sq(wc - sm.pr[v][1]) +
                         sq(c0 * SQRT4PI - sm.pr[v][2]));
      if (!(f < 3.0e38f)) f = __builtin_inff();   // NaN/Inf -> Inf
      sm.candLoss[n] = f;
    }
    __syncthreads();

    // argmin over alphas; accept only if strictly better than f
    if (tid < VPB) {
      int v = tid;
      float best = sm.candLoss[v * 8];
      int bi = 0;
      for (int ai = 1; ai < 8; ++ai) {
        float cv = sm.candLoss[v * 8 + ai];
        if (cv < best) { best = cv; bi = ai; }
      }
      sm.astep[v] = (best < sm.fcur[v]) ? exp2f(-(float)bi) : 0.0f;
    }
    __syncthreads();

    // s = a*d ; p <- p + s   (touches real region only; pads stay 0)
    for (int idx = tid; idx < VPB * NP; idx += 256) {
      int v = idx / NP, r = idx % NP;
      float s = sm.astep[v] * sm.Dv[v][r];
      sm.Sv[v][r] = s;
      sm.P[v][r] += s;
    }
    __syncthreads();

    // f1, g1 at p1
    vg_eval(sm, sm.P, sm.Gn, sm.fcur);

    // y = g1 - g ; g <- g1
    for (int idx = tid; idx < VPB * NP; idx += 256) {
      int v = idx / NP, r = idx % NP;
      float yv = sm.Gn[v][r] - sm.G[v][r];
      sm.Yv[v][r] = yv;
      sm.G[v][r]  = sm.Gn[v][r];
    }
    __syncthreads();

    // Hy = H @ y
    for (int idx = tid; idx < VPB * NP; idx += 256) {
      int v = idx / NP, r = idx % NP;
      float acc = 0.0f;
      for (int j = 0; j < NP; ++j) acc += sm.H[v][r][j] * sm.Yv[v][j];
      sm.Hy[v][r] = acc;
    }
    __syncthreads();

    // rho, y'Hy
    if (tid < VPB) {
      int v = tid;
      float sy = 0.0f, yhy = 0.0f;
      for (int j = 0; j < NP; ++j) {
        sy  += sm.Sv[v][j] * sm.Yv[v][j];
        yhy += sm.Yv[v][j] * sm.Hy[v][j];
      }
      sm.rhov[v] = (sy > 1e-10f) ? 1.0f / fmaxf(sy, 1e-10f) : 0.0f;
      sm.yhyv[v] = yhy;
    }
    __syncthreads();

    // H <- H - rho*(s Hy' + Hy s') + (rho^2 y'Hy + rho) s s'   (if rho>0)
    for (int idx = tid; idx < VPB * NP * NP; idx += 256) {
      int v = idx / (NP * NP), rc = idx % (NP * NP);
      int r = rc / NP, c = rc % NP;
      float rho = sm.rhov[v];
      if (rho > 0.0f) {
        float sr = sm.Sv[v][r], sc = sm.Sv[v][c];
        float hr = sm.Hy[v][r], hc = sm.Hy[v][c];
        float sigma = rho * rho * sm.yhyv[v] + rho;
        sm.H[v][r][c] += -rho * (sr * hc + hr * sc) + sigma * sr * sc;
      }
    }
    __syncthreads();
  }

  // ---- emit: [softplus(p0), softplus(p1), c0..c44] per voxel ----
  for (int idx = tid; idx < VPB * 47; idx += 256) {
    int v = idx / 47, j = idx % 47;
    float pv = sm.P[v][j];
    out[(size_t)(vb + v) * 47 + j] = (j < 2) ? softplusf(pv) : pv;
  }
}

extern "C" void kernel_launch(void* const* d_in, const int* in_sizes, int n_in,
                              void* d_out, int out_size, void* d_ws,
                              size_t ws_size, hipStream_t stream) {
  (void)in_sizes; (void)n_in; (void)d_ws; (void)ws_size; (void)out_size;
  const float* data   = (const float*)d_in[0];  // [16384, 64]
  const float* bvecs  = (const float*)d_in[1];  // [64, 3]
  const float* priors = (const float*)d_in[2];  // [16384, 3]
  const float* resp   = (const float*)d_in[3];  // [45]
  float* out = (float*)d_out;                   // [16384, 47]
  dim3 grid(N_VOX / VPB), block(256);
  ss3t_fit_kernel<<<grid, block, 0, stream>>>(data, bvecs, priors, resp, out);
}